// UnbatchedLSTMModel_88012469830031
// MI455X (gfx1250) — compile-verified
//
#include <hip/hip_runtime.h>
#include <hip/hip_bf16.h>

// LSTM: T=100000, IN=3, H=4 (4H = 16 gates). Serial-latency bound.
// Kernel 1: xi = x @ w_ih^T via V_WMMA_F32_16X16X4_F32 (IN padded 3->4),
//           branch-free main loop (EXEC stays all-1s through the WMMA),
//           stored gate-permuted so the scan kernel reads one float4/step.
// Kernel 2: single-wave serial scan, lane j owns hidden unit j; h broadcast
//           via v_readlane; activations via v_tanh_f32/v_exp_f32; xi loads
//           software-pipelined 8 steps ahead (L2-resident, 192 MB L2).

typedef float v2f __attribute__((ext_vector_type(2)));
typedef float v8f __attribute__((ext_vector_type(8)));

#define L2E 1.44269504088896340736f

__device__ __forceinline__ float fast_tanh(float x) {
#if __has_builtin(__builtin_amdgcn_tanhf)
  return __builtin_amdgcn_tanhf(x);
#else
  // tanh(x) = 2/(1+exp(-2x)) - 1
  float e = __builtin_amdgcn_exp2f(-2.0f * L2E * x);
  float r = __builtin_amdgcn_rcpf(1.0f + e);
  return __builtin_fmaf(2.0f, r, -1.0f);
#endif
}

__device__ __forceinline__ float fast_sigmoid(float x) {
#if __has_builtin(__builtin_amdgcn_tanhf)
  // sigmoid(x) = 0.5*tanh(x/2) + 0.5  (single trans op)
  return __builtin_fmaf(0.5f, __builtin_amdgcn_tanhf(0.5f * x), 0.5f);
#else
  float e = __builtin_amdgcn_exp2f(-L2E * x);
  return __builtin_amdgcn_rcpf(1.0f + e);
#endif
}

__device__ __forceinline__ float lane_bcast(float v, int src) {
  return __int_as_float(__builtin_amdgcn_readlane(__float_as_int(v), src));
}

// ---------------------------------------------------------------------------
// Kernel 1: xi[t][perm(n)] = sum_k x[t][k] * w_ih[n][k].
// One 16x16 tile per WMMA: A = 16 timesteps x K4, B = K4 x 16 gates, D 16x16.
// perm(n) = (n&3)*4 + (n>>2): groups (i_j, f_j, g_j, o_j) contiguous.
// A/B f32 layout: VGPR v, lane-half hh -> K = hh*2 + v.
// ---------------------------------------------------------------------------
__global__ void lstm_xi_wmma(const float* __restrict__ x,
                             const float* __restrict__ w_ih,
                             float* __restrict__ xi, int T) {
  const int lane = threadIdx.x & 31;
  const int n    = lane & 15;   // gate column (and timestep-in-tile for A)
  const int half = lane >> 4;   // K-half selector
  const int k0   = half * 2;    // 0 or 2: always a valid column index
  const int kalt = half ? 2 : 1;// clamped address for the second K element
  const int perm = (n & 3) * 4 + (n >> 2);

  // B[k][n] = w_ih[n][k]; K=3 pad lane selects 0 via cndmask (no branch).
  v2f b;
  b.x = w_ih[n * 3 + k0];
  {
    float l1 = w_ih[n * 3 + kalt];
    b.y = half ? 0.0f : l1;
  }

  const int wavesPerBlock = blockDim.x >> 5;
  const int wave = blockIdx.x * wavesPerBlock + (threadIdx.x >> 5);
  const int stride = gridDim.x * wavesPerBlock;
  const int fullTiles = T >> 4;

  for (int tile = wave; tile < fullTiles; tile += stride) {
    const int t0 = tile << 4;
    const float* xr = x + (t0 + n) * 3;
    v2f a;
    a.x = xr[k0];
    {
      float l1 = xr[kalt];              // always in-range address
      a.y = half ? 0.0f : l1;           // K=3 pad -> 0
    }
    v8f c = {};
    v8f d = __builtin_amdgcn_wmma_f32_16x16x4_f32(
        /*neg_a=*/false, a, /*neg_b=*/false, b,
        /*c_mod=*/(short)0, c, /*reuse_a=*/false, /*reuse_b=*/false);
    // D: lanes 0-15 -> M=r, N=lane ; lanes 16-31 -> M=r+8, N=lane-16.
    float* o = xi + (t0 + half * 8) * 16 + perm;
#pragma unroll
    for (int r = 0; r < 8; ++r) o[r * 16] = d[r];
  }

  // Scalar tail for T % 16 (never taken for T = 100000).
  if (wave == 0 && (T & 15)) {
    for (int t = fullTiles << 4; t < T; ++t) {
      if (lane < 16) {
        float s = 0.0f;
#pragma unroll
        for (int k = 0; k < 3; ++k)
          s = __builtin_fmaf(x[t * 3 + k], w_ih[lane * 3 + k], s);
        xi[t * 16 + perm] = s;
      }
    }
  }
}

// ---------------------------------------------------------------------------
// Kernel 2: serial scan, one wave. out = [ y(T*4) | hn(4) | cn(4) ].
// ---------------------------------------------------------------------------
__global__ void __launch_bounds__(32)
lstm_scan(const float* __restrict__ xi, const float* __restrict__ w_hh,
          float* __restrict__ out, int T) {
  const int lane = threadIdx.x & 31;
  const int j = lane & 3;  // hidden unit owned by this lane (replicated x8)

  // Rows j (i), j+4 (f), j+8 (g), j+12 (o) of w_hh (16x4 row-major).
  float wi0 = w_hh[(j     ) * 4 + 0], wi1 = w_hh[(j     ) * 4 + 1],
        wi2 = w_hh[(j     ) * 4 + 2], wi3 = w_hh[(j     ) * 4 + 3];
  float wf0 = w_hh[(j +  4) * 4 + 0], wf1 = w_hh[(j +  4) * 4 + 1],
        wf2 = w_hh[(j +  4) * 4 + 2], wf3 = w_hh[(j +  4) * 4 + 3];
  float wg0 = w_hh[(j +  8) * 4 + 0], wg1 = w_hh[(j +  8) * 4 + 1],
        wg2 = w_hh[(j +  8) * 4 + 2], wg3 = w_hh[(j +  8) * 4 + 3];
  float wo0 = w_hh[(j + 12) * 4 + 0], wo1 = w_hh[(j + 12) * 4 + 1],
        wo2 = w_hh[(j + 12) * 4 + 2], wo3 = w_hh[(j + 12) * 4 + 3];

  float h0 = 0.f, h1 = 0.f, h2 = 0.f, h3 = 0.f;
  float c = 0.f, hval = 0.f;

  const float4* __restrict__ xiv = (const float4*)xi;  // [t*4 + j]

  auto step = [&](float4 v) {
    // g = xi + w_hh @ h (h_k are SGPR-broadcast scalars)
    float gi = __builtin_fmaf(wi3, h3, __builtin_fmaf(wi2, h2,
               __builtin_fmaf(wi1, h1, __builtin_fmaf(wi0, h0, v.x))));
    float gf = __builtin_fmaf(wf3, h3, __builtin_fmaf(wf2, h2,
               __builtin_fmaf(wf1, h1, __builtin_fmaf(wf0, h0, v.y))));
    float gg = __builtin_fmaf(wg3, h3, __builtin_fmaf(wg2, h2,
               __builtin_fmaf(wg1, h1, __builtin_fmaf(wg0, h0, v.z))));
    float go = __builtin_fmaf(wo3, h3, __builtin_fmaf(wo2, h2,
               __builtin_fmaf(wo1, h1, __builtin_fmaf(wo0, h0, v.w))));
    float ai = fast_sigmoid(gi);
    float af = fast_sigmoid(gf);
    float ag = fast_tanh(gg);
    float ao = fast_sigmoid(go);
    c = __builtin_fmaf(af, c, ai * ag);
    hval = ao * fast_tanh(c);
    h0 = lane_bcast(hval, 0);
    h1 = lane_bcast(hval, 1);
    h2 = lane_bcast(hval, 2);
    h3 = lane_bcast(hval, 3);
  };

  constexpr int CH = 8;                 // steps per pipelined chunk
  const int nChunks = T / CH;
  float4 bufA[CH];
  if (nChunks > 0) {
#pragma unroll
    for (int u = 0; u < CH; ++u) bufA[u] = xiv[u * 4 + j];
  }
  for (int ch = 0; ch < nChunks; ++ch) {
    const int tbase = ch * CH;
    float4 bufN[CH];
    if (ch + 1 < nChunks) {             // prefetch next chunk (hides L2 latency)
#pragma unroll
      for (int u = 0; u < CH; ++u) bufN[u] = xiv[(tbase + CH + u) * 4 + j];
    } else {
#pragma unroll
      for (int u = 0; u < CH; ++u) bufN[u] = make_float4(0.f, 0.f, 0.f, 0.f);
    }
    float hbuf[CH];
#pragma unroll
    for (int u = 0; u < CH; ++u) {
      step(bufA[u]);
      hbuf[u] = hval;
    }
    if (lane < 4) {                     // one EXEC toggle per CH steps
#pragma unroll
      for (int u = 0; u < CH; ++u) out[(tbase + u) * 4 + j] = hbuf[u];
    }
#pragma unroll
    for (int u = 0; u < CH; ++u) bufA[u] = bufN[u];
  }
  for (int t = nChunks * CH; t < T; ++t) {  // tail (never taken for T=100000)
    step(xiv[t * 4 + j]);
    if (lane < 4) out[t * 4 + j] = hval;
  }

  if (lane < 4) {
    out[T * 4 + j] = hval;      // hn
    out[T * 4 + 4 + j] = c;     // cn
  }
}

extern "C" void kernel_launch(void* const* d_in, const int* in_sizes, int n_in,
                              void* d_out, int out_size, void* d_ws, size_t ws_size,
                              hipStream_t stream) {
  const float* x    = (const float*)d_in[0];   // (T, 3)
  const float* w_ih = (const float*)d_in[1];   // (16, 3)
  const float* w_hh = (const float*)d_in[2];   // (16, 4)
  float* out = (float*)d_out;                  // T*4 + 4 + 4
  float* xi  = (float*)d_ws;                   // T*16 floats scratch

  const int T = in_sizes[0] / 3;
  const int numTiles = (T + 15) / 16;
  const int wavesPerBlock = 8;                 // 256 threads
  const int blocks = (numTiles + wavesPerBlock - 1) / wavesPerBlock;

  lstm_xi_wmma<<<blocks, wavesPerBlock * 32, 0, stream>>>(x, w_ih, xi, T);
  lstm_scan<<<1, 32, 0, stream>>>(xi, w_hh, out, T);
}